// DynaBlock_50586124812906
// MI455X (gfx1250) — compile-verified
//
#include <hip/hip_runtime.h>
#include <hip/hip_bf16.h>

typedef __attribute__((ext_vector_type(2))) float v2f;
typedef __attribute__((ext_vector_type(8))) float v8f;

#define BQ 64
#define HQ 32
#define WQ 32
#define DQ 384

// ---------------------------------------------------------------------------
// Attention logit + softmax kernel.
// One block per (b, f) where f = w (mode 0, h-mix) or f = h (mode 1, w-mix).
// Computes y = slice @ W1^T + b1  (E=2), grouped 1x1 mix (seg=8) -> 1024
// logits, then row-softmax over 32 -> attn[(b*32+f)*1024 + i*32 + j].
// ---------------------------------------------------------------------------
__global__ __launch_bounds__(256) void attn_logits_kernel(
    const float* __restrict__ x, const float* __restrict__ W1,
    const float* __restrict__ b1, const float* __restrict__ W2,
    const float* __restrict__ b2, float* __restrict__ attn, int mode) {
  __shared__ float sX[32][DQ + 1];   // x slice, [j][d]
  __shared__ float sW1[2 * DQ];
  __shared__ float sy[64];           // y_flat[c], c = e*32 + j
  __shared__ float sZ[1024];

  const int b = blockIdx.x >> 5;
  const int f = blockIdx.x & 31;
  const size_t sj   = (mode == 0) ? (size_t)(WQ * DQ) : (size_t)DQ;
  const size_t sf   = (mode == 0) ? (size_t)DQ : (size_t)(WQ * DQ);
  const size_t base = (size_t)b * HQ * WQ * DQ + (size_t)f * sf;
  const int tid = threadIdx.x;

  for (int idx = tid; idx < 32 * DQ; idx += 256) {
    int j = idx / DQ, d = idx % DQ;
    sX[j][d] = x[base + (size_t)j * sj + d];
  }
  for (int idx = tid; idx < 2 * DQ; idx += 256) sW1[idx] = W1[idx];
  __syncthreads();

  if (tid < 64) {
    int e = tid >> 5, j = tid & 31;
    float acc = b1[e];
    #pragma unroll 8
    for (int d = 0; d < DQ; ++d) acc += sX[j][d] * sW1[e * DQ + d];
    sy[tid] = acc;
  }
  __syncthreads();

  // Grouped mix: seg=8, cin=64, cout=1024; z[ch] = b2[ch] + sum_c sy[g*8+c]*W2[ch*8+c]
  for (int ch = tid; ch < 1024; ch += 256) {
    int g = ch >> 7;
    float acc = b2[ch];
    #pragma unroll
    for (int cc = 0; cc < 8; ++cc) acc += sy[g * 8 + cc] * W2[ch * 8 + cc];
    sZ[ch] = acc;
  }
  __syncthreads();

  // Row softmax over 32 logits per row (32 rows).
  if (tid < 32) {
    const int i = tid;
    float mx = -3.0e38f;
    #pragma unroll
    for (int j = 0; j < 32; ++j) mx = fmaxf(mx, sZ[i * 32 + j]);
    float sum = 0.0f;
    #pragma unroll
    for (int j = 0; j < 32; ++j) sum += __expf(sZ[i * 32 + j] - mx);
    float inv = 1.0f / sum;
    float* outp = attn + (size_t)blockIdx.x * 1024 + i * 32;
    #pragma unroll
    for (int j = 0; j < 32; ++j) outp[j] = __expf(sZ[i * 32 + j] - mx) * inv;
  }
}

// ---------------------------------------------------------------------------
// Apply attention: per (b,f), Y_slice(32x384) += attn(32x32) @ X_slice(32x384)
// using V_WMMA_F32_16X16X4_F32. 8 waves: 2 M-tiles x 4 wave-columns x 6 N-tiles.
// ---------------------------------------------------------------------------
__global__ __launch_bounds__(256) void attn_apply_kernel(
    const float* __restrict__ x, const float* __restrict__ attn,
    float* __restrict__ Y, int mode) {
  __shared__ float sX[32][DQ + 1];   // B matrix: [k=j][n=d]
  __shared__ float sA[32][33];       // A matrix: [m=i][k=j]

  const int b = blockIdx.x >> 5;
  const int f = blockIdx.x & 31;
  const size_t sj   = (mode == 0) ? (size_t)(WQ * DQ) : (size_t)DQ;
  const size_t sf   = (mode == 0) ? (size_t)DQ : (size_t)(WQ * DQ);
  const size_t base = (size_t)b * HQ * WQ * DQ + (size_t)f * sf;
  const int tid = threadIdx.x;

  for (int idx = tid; idx < 32 * DQ; idx += 256) {
    int j = idx / DQ, d = idx % DQ;
    sX[j][d] = x[base + (size_t)j * sj + d];
  }
  const float* ap = attn + (size_t)blockIdx.x * 1024;
  for (int idx = tid; idx < 1024; idx += 256) sA[idx >> 5][idx & 31] = ap[idx];
  __syncthreads();

  const int wave = tid >> 5, lane = tid & 31;
  const int hi = lane >> 4, l15 = lane & 15;
  const int wm = (wave & 1) * 16;    // M-tile row offset: 0 or 16
  const int wg = wave >> 1;          // wave column group 0..3

  v8f acc[6];
  #pragma unroll
  for (int s = 0; s < 6; ++s) acc[s] = (v8f){0, 0, 0, 0, 0, 0, 0, 0};

  #pragma unroll
  for (int kk = 0; kk < 32; kk += 4) {
    v2f a;
    a.x = sA[wm + l15][kk + 2 * hi];
    a.y = sA[wm + l15][kk + 2 * hi + 1];
    #pragma unroll
    for (int s = 0; s < 6; ++s) {
      int n0 = (wg + 4 * s) * 16;
      v2f bf;
      bf.x = sX[kk + 2 * hi][n0 + l15];
      bf.y = sX[kk + 2 * hi + 1][n0 + l15];
      acc[s] = __builtin_amdgcn_wmma_f32_16x16x4_f32(
          false, a, false, bf, (short)0, acc[s], false, false);
    }
  }

  // Accumulate into Y (same slice geometry as x slice).
  #pragma unroll
  for (int s = 0; s < 6; ++s) {
    int n0 = (wg + 4 * s) * 16 + l15;
    #pragma unroll
    for (int r = 0; r < 8; ++r) {
      int i = wm + r + 8 * hi;
      size_t off = base + (size_t)i * sj + n0;
      Y[off] += acc[s][r];
    }
  }
}

// ---------------------------------------------------------------------------
// C[M,N] = A[M,K] @ W[N,K]^T + bias[N], fp32 WMMA 16x16x4.
// Block tile 128x64, 8 waves arranged 4(M) x 2(N); each wave owns a 32x32
// tile (4 accumulators -> 4 WMMAs per 4 LDS fragments). Double-buffered LDS
// with register staging; global_prefetch of chunk k+2 via __builtin_prefetch.
// ---------------------------------------------------------------------------
#define TM 128
#define TN 64
#define KC 32

__global__ __launch_bounds__(256) void gemm_nt_bias_kernel(
    const float* __restrict__ A, const float* __restrict__ W,
    const float* __restrict__ bias, float* __restrict__ C,
    int M, int N, int K) {
  __shared__ float sA[2][TM][KC + 1];   // 2 * 128 * 33 * 4B = 33.8 KB
  __shared__ float sB[2][TN][KC + 1];   // 2 *  64 * 33 * 4B = 16.9 KB

  const int tid = threadIdx.x;
  const int wave = tid >> 5, lane = tid & 31;
  const int hi = lane >> 4, l15 = lane & 15;
  const int bm = blockIdx.x * TM;
  const int bn = blockIdx.y * TN;
  const int wm = (wave & 3) * 32;    // 0,32,64,96
  const int wn = (wave >> 2) * 32;   // 0,32

  v8f acc00 = (v8f){0, 0, 0, 0, 0, 0, 0, 0};
  v8f acc01 = (v8f){0, 0, 0, 0, 0, 0, 0, 0};
  v8f acc10 = (v8f){0, 0, 0, 0, 0, 0, 0, 0};
  v8f acc11 = (v8f){0, 0, 0, 0, 0, 0, 0, 0};

  float ra[16];   // A staging: 128*32 / 256 threads
  float rb[8];    // B staging:  64*32 / 256 threads

  // Prologue: stage chunk 0 into buffer 0.
  #pragma unroll
  for (int i = 0; i < 16; ++i) {
    int idx = tid + i * 256, r = idx >> 5, c = idx & 31;
    ra[i] = A[(size_t)(bm + r) * K + c];
  }
  #pragma unroll
  for (int i = 0; i < 8; ++i) {
    int idx = tid + i * 256, r = idx >> 5, c = idx & 31;
    rb[i] = W[(size_t)(bn + r) * K + c];
  }
  #pragma unroll
  for (int i = 0; i < 16; ++i) {
    int idx = tid + i * 256;
    sA[0][idx >> 5][idx & 31] = ra[i];
  }
  #pragma unroll
  for (int i = 0; i < 8; ++i) {
    int idx = tid + i * 256;
    sB[0][idx >> 5][idx & 31] = rb[i];
  }
  __syncthreads();

  const int nkc = K / KC;
  for (int kc = 0; kc < nkc; ++kc) {
    const int cur = kc & 1, nxt = cur ^ 1;
    const int k1 = (kc + 1) * KC;
    const bool have_next = (k1 < K);

    if (have_next) {
      // Issue global loads for chunk kc+1 (overlaps with WMMA below).
      #pragma unroll
      for (int i = 0; i < 16; ++i) {
        int idx = tid + i * 256, r = idx >> 5, c = idx & 31;
        ra[i] = A[(size_t)(bm + r) * K + k1 + c];
        if ((i & 3) == 0 && k1 + KC < K)
          __builtin_prefetch(&A[(size_t)(bm + r) * K + k1 + KC + c], 0, 1);
      }
      #pragma unroll
      for (int i = 0; i < 8; ++i) {
        int idx = tid + i * 256, r = idx >> 5, c = idx & 31;
        rb[i] = W[(size_t)(bn + r) * K + k1 + c];
        if ((i & 3) == 0 && k1 + KC < K)
          __builtin_prefetch(&W[(size_t)(bn + r) * K + k1 + KC + c], 0, 1);
      }
    }

    // Compute current chunk: 8 K-steps x 4 WMMAs.
    #pragma unroll
    for (int kk = 0; kk < KC; kk += 4) {
      v2f a0, a1, b0, b1;
      a0.x = sA[cur][wm + l15][kk + 2 * hi];
      a0.y = sA[cur][wm + l15][kk + 2 * hi + 1];
      a1.x = sA[cur][wm + 16 + l15][kk + 2 * hi];
      a1.y = sA[cur][wm + 16 + l15][kk + 2 * hi + 1];
      b0.x = sB[cur][wn + l15][kk + 2 * hi];
      b0.y = sB[cur][wn + l15][kk + 2 * hi + 1];
      b1.x = sB[cur][wn + 16 + l15][kk + 2 * hi];
      b1.y = sB[cur][wn + 16 + l15][kk + 2 * hi + 1];
      acc00 = __builtin_amdgcn_wmma_f32_16x16x4_f32(
          false, a0, false, b0, (short)0, acc00, false, false);
      acc01 = __builtin_amdgcn_wmma_f32_16x16x4_f32(
          false, a0, false, b1, (short)0, acc01, false, false);
      acc10 = __builtin_amdgcn_wmma_f32_16x16x4_f32(
          false, a1, false, b0, (short)0, acc10, false, false);
      acc11 = __builtin_amdgcn_wmma_f32_16x16x4_f32(
          false, a1, false, b1, (short)0, acc11, false, false);
    }

    if (have_next) {
      // Store staged chunk kc+1 into the other buffer.
      #pragma unroll
      for (int i = 0; i < 16; ++i) {
        int idx = tid + i * 256;
        sA[nxt][idx >> 5][idx & 31] = ra[i];
      }
      #pragma unroll
      for (int i = 0; i < 8; ++i) {
        int idx = tid + i * 256;
        sB[nxt][idx >> 5][idx & 31] = rb[i];
      }
    }
    __syncthreads();
  }

  // Epilogue: write 32x32 wave tile with bias.
  const int nc0 = bn + wn + l15;
  const int nc1 = nc0 + 16;
  const float bv0 = bias[nc0];
  const float bv1 = bias[nc1];
  #pragma unroll
  for (int r = 0; r < 8; ++r) {
    int row0 = bm + wm + r + 8 * hi;
    int row1 = row0 + 16;
    C[(size_t)row0 * N + nc0] = acc00[r] + bv0;
    C[(size_t)row0 * N + nc1] = acc01[r] + bv1;
    C[(size_t)row1 * N + nc0] = acc10[r] + bv0;
    C[(size_t)row1 * N + nc1] = acc11[r] + bv1;
  }
}

// ---------------------------------------------------------------------------
extern "C" void kernel_launch(void* const* d_in, const int* in_sizes, int n_in,
                              void* d_out, int out_size, void* d_ws, size_t ws_size,
                              hipStream_t stream) {
  const float* x   = (const float*)d_in[0];
  const float* Wc  = (const float*)d_in[1];
  const float* bc  = (const float*)d_in[2];
  const float* Wo  = (const float*)d_in[3];
  const float* bo  = (const float*)d_in[4];
  const float* W1h = (const float*)d_in[5];
  const float* b1h = (const float*)d_in[6];
  const float* W2h = (const float*)d_in[7];
  const float* b2h = (const float*)d_in[8];
  const float* W1w = (const float*)d_in[9];
  const float* b1w = (const float*)d_in[10];
  const float* W2w = (const float*)d_in[11];
  const float* b2w = (const float*)d_in[12];
  float* out = (float*)d_out;

  const size_t M = (size_t)BQ * HQ * WQ;   // 65536
  float* Y      = (float*)d_ws;                         // [M, D] = 100.7 MB
  float* attn_h = Y + M * DQ;                           // [B*W, 32, 32] = 8.4 MB
  float* attn_w = attn_h + (size_t)BQ * WQ * 32 * 32;   // [B*H, 32, 32] = 8.4 MB

  // 1) attention weights (h-mix and w-mix)
  attn_logits_kernel<<<BQ * WQ, 256, 0, stream>>>(x, W1h, b1h, W2h, b2h, attn_h, 0);
  attn_logits_kernel<<<BQ * HQ, 256, 0, stream>>>(x, W1w, b1w, W2w, b2w, attn_w, 1);

  // 2) Y = x @ Wc^T + bc
  gemm_nt_bias_kernel<<<dim3((unsigned)(M / TM), DQ / TN), 256, 0, stream>>>(
      x, Wc, bc, Y, (int)M, DQ, DQ);

  // 3) Y += attn_h @ x-slices ; Y += attn_w @ x-slices  (serial launches)
  attn_apply_kernel<<<BQ * WQ, 256, 0, stream>>>(x, attn_h, Y, 0);
  attn_apply_kernel<<<BQ * HQ, 256, 0, stream>>>(x, attn_w, Y, 1);

  // 4) out = Y @ Wo^T + bo
  gemm_nt_bias_kernel<<<dim3((unsigned)(M / TM), DQ / TN), 256, 0, stream>>>(
      Y, Wo, bo, out, (int)M, DQ, DQ);
}